// LocationAwareAttention_49787260895464
// MI455X (gfx1250) — compile-verified
//
#include <hip/hip_runtime.h>
#include <hip/hip_bf16.h>

typedef __attribute__((ext_vector_type(2))) float v2f;
typedef __attribute__((ext_vector_type(8))) float v8f;

// ---------------------------------------------------------------------------
// Generic batched GEMM, C = A(MxK) * B(KxN), all row-major f32, via
// V_WMMA_F32_16X16X4_F32.  One 32x32 C block (2x2 WMMA tiles) per wave32:
// per k-step 2 A-frags + 2 B-frags feed 4 WMMAs (2 loads/WMMA/lane).
// M, N must be multiples of 32.
// ---------------------------------------------------------------------------
__global__ __launch_bounds__(256) void gemm_f32_wmma(
    const float* __restrict__ A, const float* __restrict__ B,
    float* __restrict__ C, int M, int N, int Kd,
    long sA, long sB, long sC) {
  int bt = blockIdx.y;
  A += (size_t)bt * sA;
  B += (size_t)bt * sB;
  C += (size_t)bt * sC;

  int wg   = blockIdx.x * 8 + (threadIdx.x >> 5);
  int lane = threadIdx.x & 31;
  int half = lane >> 4;    // A/B: K=0,1 vs K=2,3 ; D: M rows 0-7 vs 8-15
  int l15  = lane & 15;

  int tN = N >> 5;
  int mo = (wg / tN) << 5;
  int no = (wg % tN) << 5;
  if (mo >= M) return;  // wave-uniform guard (EXEC stays all-1s for WMMA)

  const float* ar0 = A + (size_t)(mo + l15) * Kd;   // A tile rows  mo..mo+15
  const float* ar1 = ar0 + (size_t)16 * Kd;         // A tile rows  mo+16..
  const float* br0 = B + no + l15;                  // B tile cols  no..no+15
  const float* br1 = br0 + 16;                      // B tile cols  no+16..

  v8f acc00 = {}, acc01 = {}, acc10 = {}, acc11 = {};
  for (int kb = 0; kb < Kd; kb += 4) {
    if ((kb & 63) == 0) {  // wave-uniform: prefetch A streams one block ahead
      __builtin_prefetch(ar0 + kb + 64, 0, 1);
      __builtin_prefetch(ar1 + kb + 64, 0, 1);
    }
    int kk = kb + (half << 1);
    v2f a0, a1, b0, b1;
    a0.x = ar0[kk];
    a0.y = ar0[kk + 1];
    a1.x = ar1[kk];
    a1.y = ar1[kk + 1];
    b0.x = br0[(size_t)kk * N];
    b0.y = br0[(size_t)(kk + 1) * N];
    b1.x = br1[(size_t)kk * N];
    b1.y = br1[(size_t)(kk + 1) * N];
    acc00 = __builtin_amdgcn_wmma_f32_16x16x4_f32(false, a0, false, b0,
                                                  (short)0, acc00, false, false);
    acc01 = __builtin_amdgcn_wmma_f32_16x16x4_f32(false, a0, false, b1,
                                                  (short)0, acc01, false, false);
    acc10 = __builtin_amdgcn_wmma_f32_16x16x4_f32(false, a1, false, b0,
                                                  (short)0, acc10, false, false);
    acc11 = __builtin_amdgcn_wmma_f32_16x16x4_f32(false, a1, false, b1,
                                                  (short)0, acc11, false, false);
  }

  // D layout: element (M = v + 8*half, N = l15) within each 16x16 tile
  float* c00 = C + (size_t)(mo + (half << 3)) * N + no + l15;
  float* c10 = c00 + (size_t)16 * N;
#pragma unroll
  for (int v = 0; v < 8; ++v) {
    c00[(size_t)v * N]      = acc00[v];
    c00[(size_t)v * N + 16] = acc01[v];
    c10[(size_t)v * N]      = acc10[v];
    c10[(size_t)v * N + 16] = acc11[v];
  }
}

// ---------------------------------------------------------------------------
// Fused location-attention energy + softmax.
// One workgroup per (b,q) row: conv(prev_attn) -> loc tile in LDS ->
// WMMA (KxF)@(FxH) -> tanh/dot(We) epilogue -> softmax -> attn out.
// B=4 Q=128 K=1024 D=512 H=512 F=32 KS=31 PAD=15, k-tile = 64.
// ---------------------------------------------------------------------------
#define AK 1024
#define AH 512
#define AF 32
#define AKS 31
#define APAD 15

__global__ __launch_bounds__(256) void energy_softmax_kernel(
    const float* __restrict__ prev, const float* __restrict__ qp,
    const float* __restrict__ kp, const float* __restrict__ Wloc,
    const float* __restrict__ convw, const float* __restrict__ convb,
    const float* __restrict__ We, float* __restrict__ attn_out) {
  const int tid  = threadIdx.x;
  const int wave = tid >> 5;
  const int lane = tid & 31;
  const int half = lane >> 4;
  const int l15  = lane & 15;
  const int bq   = blockIdx.x;       // b*Q + q
  const int b    = bq >> 7;          // Q = 128

  __shared__ float s_pa[AK + 2 * APAD];     // zero-padded prev-attn row
  __shared__ float s_qp[AH];
  __shared__ float s_we[AH];
  __shared__ float s_cw[AF * AKS];
  __shared__ float s_cb[AF];
  __shared__ float s_loc[64][AF + 1];       // loc tile [k_local][f], padded
  __shared__ float s_epart[8][AK];          // per-wave energy slices
  __shared__ float s_energy[AK];
  __shared__ float s_red[8];
  __shared__ float s_red2[8];

  // ---- stage constants / row data into LDS -------------------------------
  const float* parow = prev + (size_t)bq * AK;
  for (int i = tid; i < APAD; i += 256) {
    s_pa[i] = 0.f;
    s_pa[APAD + AK + i] = 0.f;
  }
  for (int i = tid; i < AK; i += 256) s_pa[APAD + i] = parow[i];
  for (int i = tid; i < AH; i += 256) {
    s_qp[i] = qp[(size_t)bq * AH + i];
    s_we[i] = We[i];
  }
  for (int i = tid; i < AF * AKS; i += 256) s_cw[i] = convw[i];
  for (int i = tid; i < AF; i += 256) s_cb[i] = convb[i];

  // ---- hoist B fragments of Wloc (constant for whole kernel) -------------
  // wave handles h-tiles {wave, wave+8, wave+16, wave+24}
  v2f bfr[4][8];
#pragma unroll
  for (int hi = 0; hi < 4; ++hi) {
    int ho = ((hi << 3) + wave) << 4;
#pragma unroll
    for (int fq = 0; fq < 8; ++fq) {
      int kk = (fq << 2) + (half << 1);
      bfr[hi][fq].x = Wloc[(size_t)kk * AH + ho + l15];
      bfr[hi][fq].y = Wloc[(size_t)(kk + 1) * AH + ho + l15];
    }
  }
  __syncthreads();

  // ---- main loop over k-tiles of 64 --------------------------------------
  for (int kt = 0; kt < AK / 64; ++kt) {
    int k0 = kt << 6;

    // conv: loc[f,k] = sum_t w[f,t] * pa[k+t-15] + b[f]   (64x32 tile)
    for (int e = tid; e < 64 * AF; e += 256) {
      int kl = e >> 5;
      int f  = e & (AF - 1);
      const float* w = &s_cw[f * AKS];
      const float* p = &s_pa[k0 + kl];
      float s = s_cb[f];
#pragma unroll
      for (int t = 0; t < AKS; ++t) s = fmaf(w[t], p[t], s);
      s_loc[kl][f] = s;
    }
    __syncthreads();

    // WMMA: (16k x 32f) @ (32f x 16h) tiles + tanh/dot epilogue
#pragma unroll 1
    for (int ks = 0; ks < 4; ++ks) {
      v2f afr[8];
#pragma unroll
      for (int fq = 0; fq < 8; ++fq) {
        int kk = (fq << 2) + (half << 1);
        afr[fq].x = s_loc[(ks << 4) + l15][kk];
        afr[fq].y = s_loc[(ks << 4) + l15][kk + 1];
      }
      float esum[8] = {0.f, 0.f, 0.f, 0.f, 0.f, 0.f, 0.f, 0.f};
#pragma unroll 1
      for (int hi = 0; hi < 4; ++hi) {
        int ho = ((hi << 3) + wave) << 4;
        v8f acc = {};
#pragma unroll
        for (int fq = 0; fq < 8; ++fq)
          acc = __builtin_amdgcn_wmma_f32_16x16x4_f32(
              false, afr[fq], false, bfr[hi][fq], (short)0, acc, false, false);
        int h = ho + l15;
        float wh  = s_we[h];
        float qph = s_qp[h];
        const float* kprow =
            kp + ((size_t)b * AK + k0 + (ks << 4) + (half << 3)) * AH + h;
#pragma unroll
        for (int v = 0; v < 8; ++v) {
          float x = acc[v] + qph + kprow[(size_t)v * AH];
          x = fminf(fmaxf(x, -15.f), 15.f);
          float t = __expf(2.f * x);                 // tanh via v_exp_f32
          esum[v] = fmaf(wh, (t - 1.f) / (t + 1.f), esum[v]);
        }
      }
      // reduce over N (16 lanes) -> one energy value per k row
#pragma unroll
      for (int v = 0; v < 8; ++v) {
        float val = esum[v];
#pragma unroll
        for (int m = 1; m < 16; m <<= 1) val += __shfl_xor(val, m, 32);
        if (l15 == 0)
          s_epart[wave][k0 + (ks << 4) + (half << 3) + v] = val;
      }
    }
    __syncthreads();
  }

  // ---- softmax over k (deterministic tree combine of wave slices) --------
  float lmax = -3.402823466e38f;
  for (int i = tid; i < AK; i += 256) {
    float s = 0.f;
#pragma unroll
    for (int w2 = 0; w2 < 8; ++w2) s += s_epart[w2][i];
    s_energy[i] = s;
    lmax = fmaxf(lmax, s);
  }
#pragma unroll
  for (int m = 16; m >= 1; m >>= 1) lmax = fmaxf(lmax, __shfl_xor(lmax, m, 32));
  if (lane == 0) s_red[wave] = lmax;
  __syncthreads();
  float gmax = s_red[0];
#pragma unroll
  for (int w2 = 1; w2 < 8; ++w2) gmax = fmaxf(gmax, s_red[w2]);

  float lsum = 0.f;
  for (int i = tid; i < AK; i += 256) {
    float e = __expf(s_energy[i] - gmax);
    s_energy[i] = e;
    lsum += e;
  }
#pragma unroll
  for (int m = 16; m >= 1; m >>= 1) lsum += __shfl_xor(lsum, m, 32);
  if (lane == 0) s_red2[wave] = lsum;
  __syncthreads();
  float gsum = 0.f;
#pragma unroll
  for (int w2 = 0; w2 < 8; ++w2) gsum += s_red2[w2];
  float inv = 1.f / gsum;

  float* arow = attn_out + (size_t)bq * AK;
  for (int i = tid; i < AK; i += 256) arow[i] = s_energy[i] * inv;
}

// ---------------------------------------------------------------------------
// Launch: qp/kp GEMMs -> fused energy+softmax -> context GEMM.
// ---------------------------------------------------------------------------
extern "C" void kernel_launch(void* const* d_in, const int* in_sizes, int n_in,
                              void* d_out, int out_size, void* d_ws,
                              size_t ws_size, hipStream_t stream) {
  const float* query = (const float*)d_in[0];  // (4,128,512)
  const float* key   = (const float*)d_in[1];  // (4,1024,512)
  const float* value = (const float*)d_in[2];  // (4,1024,512)
  const float* prev  = (const float*)d_in[3];  // (4,128,1024)
  const float* Wq    = (const float*)d_in[4];  // (512,512)
  const float* Wk    = (const float*)d_in[5];  // (512,512)
  const float* convw = (const float*)d_in[6];  // (32,1,31)
  const float* convb = (const float*)d_in[7];  // (32,)
  const float* Wloc  = (const float*)d_in[8];  // (32,512)
  const float* We    = (const float*)d_in[9];  // (512,)

  float* context = (float*)d_out;              // (4,128,512) = 262144 floats
  float* attn    = context + 4 * 128 * 512;    // (4,128,1024) = 524288 floats

  float* qp = (float*)d_ws;                    // (512,512)
  float* kp = qp + 512 * 512;                  // (4096,512)

  // qp = query @ Wq : M=512 N=512 -> 16x16 blocks of 32x32 = 256 waves
  gemm_f32_wmma<<<dim3(32, 1), 256, 0, stream>>>(query, Wq, qp, 512, 512, 512,
                                                 0, 0, 0);
  // kp = key @ Wk : M=4096 N=512 -> 128x16 = 2048 waves -> 256 blocks
  gemm_f32_wmma<<<dim3(256, 1), 256, 0, stream>>>(key, Wk, kp, 4096, 512, 512,
                                                  0, 0, 0);
  // fused conv + loc-proj + energy + softmax: one WG per (b,q)
  energy_softmax_kernel<<<dim3(512), 256, 0, stream>>>(prev, qp, kp, Wloc,
                                                       convw, convb, We, attn);
  // context = attn @ value (batched over B): per-b 4x16 = 64 waves -> 8 blocks
  gemm_f32_wmma<<<dim3(8, 4), 256, 0, stream>>>(
      attn, value, context, 128, 512, 1024, 128L * 1024, 1024L * 512,
      128L * 512);
}